// Aggregator_22763326668902
// MI455X (gfx1250) — compile-verified
//
#include <hip/hip_runtime.h>

// D = X(16x64) @ W^T(64x64) done with V_WMMA_F32_16X16X4_F32 (exact fp32 path).
typedef float v2f __attribute__((ext_vector_type(2)));
typedef float v8f __attribute__((ext_vector_type(8)));

#define K_NEI 32
#define DIM 64
#define ROWS_PER_WAVE 16
#define WAVES_PER_BLOCK 8
#define ROWS_PER_BLOCK (ROWS_PER_WAVE * WAVES_PER_BLOCK)
#define LDS_STRIDE 68   // 16B-aligned rows, conflict-free b64 reads in A-layout

__global__ __launch_bounds__(256)
void aggregator_gnn_kernel(const float* __restrict__ self_v,
                           const float* __restrict__ nei_v,
                           const float* __restrict__ rel_v,
                           const float* __restrict__ norms,
                           const float* __restrict__ user_e,
                           const float* __restrict__ W,
                           const float* __restrict__ bias,
                           float* __restrict__ out,
                           int Bmask) {
    __shared__ __align__(16) float Xl[WAVES_PER_BLOCK][ROWS_PER_WAVE][LDS_STRIDE];

    const int tid    = threadIdx.x;
    const int wv     = tid >> 5;
    const int lane   = tid & 31;        // wave32 lane
    const int part   = lane >> 4;       // 0: lanes 0-15, 1: lanes 16-31
    const int lane16 = lane & 15;
    const int rowBase = blockIdx.x * ROWS_PER_BLOCK + wv * ROWS_PER_WAVE;

    // ---------------- Phase 1: per-row score -> softmax -> weighted agg ----
    for (int r = 0; r < ROWS_PER_WAVE; ++r) {
        const int row = rowBase + r;
        const size_t nb = (size_t)row * (K_NEI * DIM);
        const float* relRow  = rel_v + nb;
        const float* neiRow  = nei_v + nb;
        const float* userRow = user_e + (size_t)(row & Bmask) * DIM;

        // Prefetch this row's neighbor vectors (8 KB: 32 lanes x 256B apart)
        __builtin_prefetch(neiRow + (size_t)lane * 64, 0, 0);

        // lane = neighbor index k: score_k = rel[r,k,:] . user[r,:]
        float score = 0.0f;
        const float4* relK = (const float4*)(relRow + (size_t)lane * DIM);
        const float4* uv4  = (const float4*)userRow;   // wave-uniform -> SMEM
        #pragma unroll
        for (int d4 = 0; d4 < DIM / 4; ++d4) {
            float4 a = relK[d4];
            float4 u = uv4[d4];
            score = fmaf(a.x, u.x, fmaf(a.y, u.y,
                    fmaf(a.z, u.z, fmaf(a.w, u.w, score))));
        }

        // wave32 softmax over K=32 neighbors
        float m = score;
        #pragma unroll
        for (int off = 16; off > 0; off >>= 1)
            m = fmaxf(m, __shfl_xor(m, off));
        float e = __expf(score - m);
        float s = e;
        #pragma unroll
        for (int off = 16; off > 0; off >>= 1)
            s += __shfl_xor(s, off);
        // fold norm and the 1/K mean into one coefficient held by lane k
        float coef = (e / s) * norms[(size_t)row * K_NEI + lane] * (1.0f / K_NEI);

        // lane role switch: stream nei[r,:,:] coalesced, 2 k-rows per b128 wave-load
        float4 acc = make_float4(0.f, 0.f, 0.f, 0.f);
        #pragma unroll
        for (int t = 0; t < 16; ++t) {
            float4 v = *(const float4*)(neiRow + (size_t)t * 128 + (size_t)lane * 4);
            int ksrc = 2 * t + part;            // which k this half-wave carries
            float c = __shfl(coef, ksrc);       // ds_bpermute broadcast
            acc.x = fmaf(v.x, c, acc.x);
            acc.y = fmaf(v.y, c, acc.y);
            acc.z = fmaf(v.z, c, acc.z);
            acc.w = fmaf(v.w, c, acc.w);
        }
        // combine even-k half (lanes 0-15) with odd-k half (lanes 16-31)
        acc.x += __shfl_xor(acc.x, 16);
        acc.y += __shfl_xor(acc.y, 16);
        acc.z += __shfl_xor(acc.z, 16);
        acc.w += __shfl_xor(acc.w, 16);

        if (lane < 16) {   // lane l owns dims 4l..4l+3 of the aggregated row
            float4 sv = *(const float4*)(self_v + (size_t)row * DIM + lane * 4);
            float4 x;
            x.x = acc.x + sv.x;
            x.y = acc.y + sv.y;
            x.z = acc.z + sv.z;
            x.w = acc.w + sv.w;
            *(float4*)&Xl[wv][r][lane * 4] = x;   // stage X tile for WMMA
        }
    }

    __syncthreads();

    // ---------------- Phase 2: relu(X @ W^T + b) via V_WMMA_F32_16X16X4 ----
    // A 16x4 layout: v0 = A[M=lane16][kk4 + 2*part], v1 = next K (ISA 7.12.2)
    // B 4x16 layout: v0 = W^T[kk4 + 2*part][n] = W[n][kk4 + 2*part], v1 = next K
    for (int jt = 0; jt < 4; ++jt) {
        v8f acc = {0.f, 0.f, 0.f, 0.f, 0.f, 0.f, 0.f, 0.f};
        const int n = jt * 16 + lane16;
        #pragma unroll
        for (int kk = 0; kk < 16; ++kk) {
            const int col = kk * 4 + part * 2;
            float2 av = *(const float2*)&Xl[wv][lane16][col];       // ds_load_b64
            float2 bw = *(const float2*)(W + (size_t)n * DIM + col); // hot in L2
            v2f a; a.x = av.x; a.y = av.y;
            v2f bv; bv.x = bw.x; bv.y = bw.y;
            acc = __builtin_amdgcn_wmma_f32_16x16x4_f32(
                      false, a, false, bv, (short)0, acc, false, false);
        }
        // C/D layout: VGPR i -> M = i + 8*part, N = lane16
        const float bb = bias[n];
        #pragma unroll
        for (int i = 0; i < 8; ++i) {
            const int mrow = i + part * 8;
            float o = acc[i] + bb;
            o = fmaxf(o, 0.0f);
            out[(size_t)(rowBase + mrow) * DIM + n] = o;
        }
    }
}

extern "C" void kernel_launch(void* const* d_in, const int* in_sizes, int n_in,
                              void* d_out, int out_size, void* d_ws, size_t ws_size,
                              hipStream_t stream) {
    const float* self_v = (const float*)d_in[0];
    const float* nei_v  = (const float*)d_in[1];
    const float* rel_v  = (const float*)d_in[2];
    const float* norms  = (const float*)d_in[3];
    const float* user_e = (const float*)d_in[4];
    const float* W      = (const float*)d_in[5];
    const float* bias   = (const float*)d_in[6];
    float* out = (float*)d_out;

    const int R = in_sizes[0] / DIM;   // 65536
    const int B = in_sizes[4] / DIM;   // 4096 (power of two)
    const int grid = R / ROWS_PER_BLOCK;

    aggregator_gnn_kernel<<<grid, 256, 0, stream>>>(
        self_v, nei_v, rel_v, norms, user_e, W, bias, out, B - 1);
}